// Decoder_66408784331209
// MI455X (gfx1250) — compile-verified
//
#include <hip/hip_runtime.h>
#include <hip/hip_bf16.h>
#include <math.h>

// ---------------- problem constants (match reference) ----------------
constexpr int Bb  = 4;
constexpr int Ss  = 2048;
constexpr int Dd  = 1024;
constexpr int Hh  = 16;
constexpr int HDd = 64;       // head dim
constexpr int FFf = 4096;
constexpr int MS  = Bb * Ss;  // 8192 activation rows

typedef __attribute__((ext_vector_type(2))) float v2f;
typedef __attribute__((ext_vector_type(8))) float v8f;

__device__ __forceinline__ v8f wmma_f32(v2f a, v2f b, v8f c) {
    // D = A(16x4) * B(4x16) + C(16x16), fp32 everywhere.
    return __builtin_amdgcn_wmma_f32_16x16x4_f32(false, a, false, b, (short)0, c,
                                                 false, false);
}

__device__ __forceinline__ float gelu_tanh(float x) {
    const float c = 0.79788456080286535588f; // sqrt(2/pi)
    return 0.5f * x * (1.0f + tanhf(c * (x + 0.044715f * x * x * x)));
}

// ---- CDNA5 async copy: global -> LDS (ASYNCcnt), 16B per lane ----
__device__ __forceinline__ void async_copy_b128(unsigned lds_off, const void* gptr) {
    asm volatile("global_load_async_to_lds_b128 %0, %1, off"
                 :: "v"(lds_off), "v"((unsigned long long)(uintptr_t)gptr)
                 : "memory");
}
__device__ __forceinline__ void wait_async0() {          // all copies done
    asm volatile("s_wait_asynccnt 0" ::: "memory");
}
__device__ __forceinline__ void wait_async6() {          // prev chunk done (in-order)
    asm volatile("s_wait_asynccnt 6" ::: "memory");
}

// ---- VALU half-wave (16-lane) butterfly reductions via DPP ----------
#define DPP8_SEL(a,b,c,d,e,f,g,h) \
    ((a)|((b)<<3)|((c)<<6)|((d)<<9)|((e)<<12)|((f)<<15)|((g)<<18)|((h)<<21))

template <int IMM>
__device__ __forceinline__ float dpp8f(float x) {
    return __builtin_bit_cast(float,
        __builtin_amdgcn_mov_dpp8(__builtin_bit_cast(int, x), IMM));
}
__device__ __forceinline__ float dpp_xor8(float x) {   // row_ror:8 == xor 8 in 16-lane row
    return __builtin_bit_cast(float,
        __builtin_amdgcn_update_dpp(0, __builtin_bit_cast(int, x),
                                    0x128 /*ROW_ROR:8*/, 0xf, 0xf, true));
}
__device__ __forceinline__ float half16_max(float v) {
    v = fmaxf(v, dpp8f<DPP8_SEL(1,0,3,2,5,4,7,6)>(v));   // xor 1
    v = fmaxf(v, dpp8f<DPP8_SEL(2,3,0,1,6,7,4,5)>(v));   // xor 2
    v = fmaxf(v, dpp8f<DPP8_SEL(4,5,6,7,0,1,2,3)>(v));   // xor 4
    v = fmaxf(v, dpp_xor8(v));                           // xor 8
    return v;
}
__device__ __forceinline__ float half16_sum(float v) {
    v += dpp8f<DPP8_SEL(1,0,3,2,5,4,7,6)>(v);
    v += dpp8f<DPP8_SEL(2,3,0,1,6,7,4,5)>(v);
    v += dpp8f<DPP8_SEL(4,5,6,7,0,1,2,3)>(v);
    v += dpp_xor8(v);
    return v;
}

// ---------------------------------------------------------------------
// LDS-tiled GEMM: C[M,N] = act( A[M,K] @ W[N,K]^T + bias [+ residual] )
// Block = 256 threads (8 waves) owns a 64x128 macro-tile.
// Double-buffered: chunk c+1 is staged to LDS with
// global_load_async_to_lds_b128 while chunk c feeds the WMMA pipe.
// ASYNC loads complete in order, 6 per thread per chunk, so
// s_wait_asynccnt 6 after issuing chunk c+1 guarantees chunk c landed.
// ---------------------------------------------------------------------
constexpr int BM = 64, BN = 128, KC = 32;
constexpr int LDA = KC + 4;   // 36 floats: 16B-aligned rows, conflict-free frags

template <bool SPLIT_HEADS, bool RESID, bool GELU_ACT>
__global__ void wgemm_kernel(const float* __restrict__ A,
                             const float* __restrict__ W,
                             const float* __restrict__ bias,
                             const float* __restrict__ R,
                             float* __restrict__ C,
                             int M, int N, int K) {
    __shared__ float sA[2][BM * LDA];   // 2 x  9216 B
    __shared__ float sB[2][BN * LDA];   // 2 x 18432 B
    const int tid  = threadIdx.x;
    const int lane = tid & 31;
    const int wave = tid >> 5;
    const int bn0  = blockIdx.x * BN;
    const int bm0  = blockIdx.y * BM;
    const int wr   = wave >> 1;            // 0..3 row tile
    const int wc   = wave & 1;             // 0..1 col half (64 cols)
    const int lh   = lane & 15;
    const int koff = (lane >> 4) << 1;     // 0 / 2
    const int half8 = (lane >> 4) << 3;    // 0 / 8

    // issue the 6 async copies (2 for A, 4 for W) of one K-chunk
    auto stage_chunk = [&](int buf, int k0) {
        const unsigned abase = (unsigned)(uintptr_t)(&sA[buf][0]);
        const unsigned bbase = (unsigned)(uintptr_t)(&sB[buf][0]);
        const int r0 = tid >> 3,          c0 = (tid & 7) * 4;
        const int r1 = (tid + 256) >> 3,  c1 = ((tid + 256) & 7) * 4;
        async_copy_b128(abase + (unsigned)(r0 * LDA + c0) * 4,
                        A + (size_t)(bm0 + r0) * K + k0 + c0);
        async_copy_b128(abase + (unsigned)(r1 * LDA + c1) * 4,
                        A + (size_t)(bm0 + r1) * K + k0 + c1);
#pragma unroll
        for (int j = 0; j < 4; ++j) {
            const int c  = tid + 256 * j;
            const int rr = c >> 3, cc = (c & 7) * 4;
            async_copy_b128(bbase + (unsigned)(rr * LDA + cc) * 4,
                            W + (size_t)(bn0 + rr) * K + k0 + cc);
        }
    };

    v8f acc[4] = {{}, {}, {}, {}};
    const int NC = K / KC;

    stage_chunk(0, 0);
    for (int c = 0; c < NC; ++c) {
        if (c + 1 < NC) {            // prefetch next chunk into other buffer
            stage_chunk((c + 1) & 1, (c + 1) * KC);
            wait_async6();           // chunk c arrived (in-order completion)
        } else {
            wait_async0();
        }
        __syncthreads();

        const float* __restrict__ ap = &sA[c & 1][(wr * 16 + lh) * LDA + koff];
        const float* __restrict__ bp = &sB[c & 1][(wc * 64 + lh) * LDA + koff];
#pragma unroll
        for (int ks = 0; ks < KC / 4; ++ks) {
            v2f a  = *(const v2f*)(ap + ks * 4);
            v2f b0 = *(const v2f*)(bp + ks * 4);
            v2f b1 = *(const v2f*)(bp + 16 * LDA + ks * 4);
            v2f b2 = *(const v2f*)(bp + 32 * LDA + ks * 4);
            v2f b3 = *(const v2f*)(bp + 48 * LDA + ks * 4);
            acc[0] = wmma_f32(a, b0, acc[0]);
            acc[1] = wmma_f32(a, b1, acc[1]);
            acc[2] = wmma_f32(a, b2, acc[2]);
            acc[3] = wmma_f32(a, b3, acc[3]);
        }
        __syncthreads();
    }

    // ---- epilogue: bias [+residual] [+gelu], optional split-head store ----
    const int row0 = bm0 + wr * 16 + half8;
#pragma unroll
    for (int t = 0; t < 4; ++t) {
        const int colg = bn0 + wc * 64 + t * 16 + lh;
        const float bv = bias[colg];
#pragma unroll
        for (int i = 0; i < 8; ++i) {
            const int rg = row0 + i;
            float v = acc[t][i] + bv;
            if (RESID)    v += R[(size_t)rg * N + colg];
            if (GELU_ACT) v  = gelu_tanh(v);
            if (SPLIT_HEADS) {
                const int b_ = rg / Ss, s_ = rg - b_ * Ss;
                const int h  = colg >> 6, hd = colg & 63;
                C[(((size_t)b_ * Hh + h) * Ss + s_) * HDd + hd] = v;
            } else {
                C[(size_t)rg * N + colg] = v;
            }
        }
    }
}

// ---------------------------------------------------------------------
// Flash-style causal attention. One wave owns one 16-row Q tile of one
// (b,h). Pipelined per 16-key tile: V loads issue before the score
// WMMA chain; next tile's K fragments prefetch behind the softmax and
// P@V WMMAs. Softmax reductions are pure-VALU DPP butterflies.
// ---------------------------------------------------------------------
__global__ void attn_kernel(const float* __restrict__ Q,
                            const float* __restrict__ Km,
                            const float* __restrict__ V,
                            float* __restrict__ O) {
    __shared__ float sm[4][16][18];          // per-wave P tile, b64-aligned pad
    const int lane = threadIdx.x & 31;
    const int wave = threadIdx.x >> 5;
    const int qt   = blockIdx.x * 4 + wave;
    const int bh   = blockIdx.y;
    const int qbase = qt * 16;
    const int lh    = lane & 15;
    const int koff  = (lane >> 4) << 1;      // 0 / 2
    const int half8 = (lane >> 4) << 3;      // 0 / 8

    // preload & pre-scale Q tile as 16 A-fragments (k-step 4)
    const float* __restrict__ Qp = Q + ((size_t)bh * Ss + qbase) * HDd
                                     + (size_t)lh * HDd + koff;
    v2f qa[16];
#pragma unroll
    for (int i = 0; i < 16; ++i) qa[i] = *(const v2f*)(Qp + i * 4) * 0.125f;

    v8f o[4] = {{}, {}, {}, {}};
    float mrow[8], lrow[8];
#pragma unroll
    for (int i = 0; i < 8; ++i) { mrow[i] = -__builtin_inff(); lrow[i] = 0.f; }

    const size_t frag_off = (size_t)lh * HDd + koff;
    const float* __restrict__ Kbh = Km + (size_t)bh * Ss * HDd;
    const float* __restrict__ Vbh = V  + (size_t)bh * Ss * HDd;

    // prime the K-fragment pipeline with tile 0
    v2f kf[16];
#pragma unroll
    for (int i = 0; i < 16; ++i) kf[i] = *(const v2f*)(Kbh + frag_off + i * 4);

    const int kt_end = qbase >> 4;
    for (int kt = 0; kt <= kt_end; ++kt) {
        const int kb = kt * 16;

        // ---- batched V-fragment loads (hide behind the score WMMAs) ----
        const float* __restrict__ Vp = Vbh + (size_t)kb * HDd;
        v2f vb[16];
#pragma unroll
        for (int j = 0; j < 4; ++j) {
            const float* __restrict__ Vr = Vp + (size_t)(j * 4 + koff) * HDd;
#pragma unroll
            for (int t = 0; t < 4; ++t) {
                vb[j * 4 + t].x = Vr[t * 16 + lh];
                vb[j * 4 + t].y = Vr[HDd + t * 16 + lh];
            }
        }

        // ---- scores: S = Qs @ K^T ----
        v8f s = {};
#pragma unroll
        for (int i = 0; i < 16; ++i) s = wmma_f32(qa[i], kf[i], s);

        // ---- prefetch next K tile (hides behind softmax + P@V) ----
        if (kt < kt_end) {
            const float* __restrict__ Kp = Kbh + (size_t)(kb + 16) * HDd + frag_off;
#pragma unroll
            for (int i = 0; i < 16; ++i) kf[i] = *(const v2f*)(Kp + i * 4);
        }

        // ---- causal mask + online softmax (DPP reductions, no DS) ----
        const int col = kb + lh;
#pragma unroll
        for (int i = 0; i < 8; ++i) {
            const int rowg = qbase + half8 + i;
            float v = s[i];
            if (col > rowg) v = -__builtin_inff();
            const float mnew = fmaxf(mrow[i], half16_max(v));
            const float sc = __expf(mrow[i] - mnew);
            const float pe = __expf(v - mnew);
            lrow[i] = lrow[i] * sc + half16_sum(pe);
            mrow[i] = mnew;
            o[0][i] *= sc; o[1][i] *= sc; o[2][i] *= sc; o[3][i] *= sc;
            sm[wave][half8 + i][lh] = pe;    // stash P in D-layout
        }

        // ---- O += P @ V : P re-read in A-layout from LDS ----
#pragma unroll
        for (int j = 0; j < 4; ++j) {
            v2f a = *(const v2f*)&sm[wave][lh][j * 4 + koff];
#pragma unroll
            for (int t = 0; t < 4; ++t)
                o[t] = wmma_f32(a, vb[j * 4 + t], o[t]);
        }
    }

    // ---- normalize, write merged-head output [B,S,D] ----
    const int b_ = bh / Hh, h = bh - b_ * Hh;
#pragma unroll
    for (int i = 0; i < 8; ++i) {
        const float inv = 1.0f / lrow[i];
        const int rg = qbase + half8 + i;
        float* __restrict__ Orow = O + ((size_t)b_ * Ss + rg) * Dd + h * HDd;
#pragma unroll
        for (int t = 0; t < 4; ++t) Orow[t * 16 + lh] = o[t][i] * inv;
    }
}

// ---------------------------------------------------------------------
// Row LayerNorm: one 256-thread block per row of N=1024.
// ---------------------------------------------------------------------
__global__ void ln_kernel(const float* __restrict__ X,
                          const float* __restrict__ g,
                          const float* __restrict__ be,
                          float* __restrict__ Y, int N) {
    const int row = blockIdx.x;
    const float* __restrict__ x = X + (size_t)row * N;
    float s = 0.f, s2 = 0.f;
    for (int i = threadIdx.x; i < N; i += blockDim.x) {
        const float v = x[i];
        s += v; s2 += v * v;
    }
#pragma unroll
    for (int m = 16; m >= 1; m >>= 1) {
        s  += __shfl_xor(s,  m, 32);
        s2 += __shfl_xor(s2, m, 32);
    }
    __shared__ float rs[8], rs2[8];
    const int wave = threadIdx.x >> 5, lane = threadIdx.x & 31;
    if (lane == 0) { rs[wave] = s; rs2[wave] = s2; }
    __syncthreads();
    if (wave == 0) {
        s  = (lane < 8) ? rs[lane]  : 0.f;
        s2 = (lane < 8) ? rs2[lane] : 0.f;
#pragma unroll
        for (int m = 4; m >= 1; m >>= 1) {
            s  += __shfl_xor(s,  m, 32);
            s2 += __shfl_xor(s2, m, 32);
        }
        if (lane == 0) { rs[0] = s; rs2[0] = s2; }
    }
    __syncthreads();
    const float mean = rs[0] / N;
    const float var  = rs2[0] / N - mean * mean;
    const float rstd = rsqrtf(var + 1e-6f);
    float* __restrict__ y = Y + (size_t)row * N;
    for (int i = threadIdx.x; i < N; i += blockDim.x)
        y[i] = g[i] * (x[i] - mean) * rstd + be[i];
}

// ---------------------------------------------------------------------
extern "C" void kernel_launch(void* const* d_in, const int* in_sizes, int n_in,
                              void* d_out, int out_size, void* d_ws, size_t ws_size,
                              hipStream_t stream) {
    const float* x  = (const float*)d_in[0];
    const float* Wq = (const float*)d_in[1];  const float* bq = (const float*)d_in[2];
    const float* Wk = (const float*)d_in[3];  const float* bk = (const float*)d_in[4];
    const float* Wv = (const float*)d_in[5];  const float* bv = (const float*)d_in[6];
    const float* Wo = (const float*)d_in[7];  const float* bo = (const float*)d_in[8];
    const float* g1 = (const float*)d_in[9];  const float* be1 = (const float*)d_in[10];
    const float* W1 = (const float*)d_in[11]; const float* b1  = (const float*)d_in[12];
    const float* W2 = (const float*)d_in[13]; const float* b2  = (const float*)d_in[14];
    const float* g2 = (const float*)d_in[15]; const float* be2 = (const float*)d_in[16];
    float* out = (float*)d_out;

    // workspace carve-up (floats); buffers reused once their producer is dead
    const size_t act = (size_t)MS * Dd;          // 8M floats per activation buf
    float* ws   = (float*)d_ws;
    float* Qb   = ws + 0 * act;                  // Q [B,H,S,HD]
    float* Kb   = ws + 1 * act;                  // K
    float* Vb   = ws + 2 * act;                  // V
    float* AO   = ws + 3 * act;                  // attention out, merged heads
    float* res1 = Qb;                            // x + attn proj   (Q dead)
    float* hbuf = Kb;                            // LN1 output      (K dead)
    float* gbuf = ws + 4 * act;                  // GELU(h@W1^T+b1) [MS,FF]
    float* res2 = Vb;                            // h + ffn         (V dead)

    const int threads = 256;
    const dim3 gD (Dd  / BN, MS / BM);           // (8, 128)
    const dim3 gFF(FFf / BN, MS / BM);           // (32, 128)

    // 1) QKV projections, split-head outputs
    wgemm_kernel<true, false, false><<<gD, threads, 0, stream>>>(x, Wq, bq, nullptr, Qb, MS, Dd, Dd);
    wgemm_kernel<true, false, false><<<gD, threads, 0, stream>>>(x, Wk, bk, nullptr, Kb, MS, Dd, Dd);
    wgemm_kernel<true, false, false><<<gD, threads, 0, stream>>>(x, Wv, bv, nullptr, Vb, MS, Dd, Dd);

    // 2) causal flash attention (4 waves / block, wave per 16-row q tile)
    attn_kernel<<<dim3(Ss / 16 / 4, Bb * Hh), 128, 0, stream>>>(Qb, Kb, Vb, AO);

    // 3) output projection + bias + residual(x)
    wgemm_kernel<false, true, false><<<gD, threads, 0, stream>>>(AO, Wo, bo, x, res1, MS, Dd, Dd);

    // 4) LayerNorm 1
    ln_kernel<<<MS, threads, 0, stream>>>(res1, g1, be1, hbuf, Dd);

    // 5) FFN up-projection + GELU
    wgemm_kernel<false, false, true><<<gFF, threads, 0, stream>>>(hbuf, W1, b1, nullptr, gbuf, MS, FFf, Dd);

    // 6) FFN down-projection + bias + residual(h)
    wgemm_kernel<false, true, false><<<gD, threads, 0, stream>>>(gbuf, W2, b2, hbuf, res2, MS, Dd, FFf);

    // 7) LayerNorm 2 -> final output
    ln_kernel<<<MS, threads, 0, stream>>>(res2, g2, be2, out, Dd);
}